// HDMD_15083925144366
// MI455X (gfx1250) — compile-verified
//
#include <hip/hip_runtime.h>
#include <math.h>

// ---------------------------------------------------------------------------
// Problem constants (from reference)
// ---------------------------------------------------------------------------
#define BB    256
#define TT    64
#define PHYS  3
#define LAT   8
#define NN    128
#define NHID  3
#define WINW  32
#define PRED  64
#define MROWS (BB * TT)          // 16384
#define RR    33                 // T - (WIN-1)
#define NSV   31                 // WIN-1 columns of Ym

typedef float v2f __attribute__((ext_vector_type(2)));
typedef float v8f __attribute__((ext_vector_type(8)));

__device__ __forceinline__ int imin(int a, int b) { return a < b ? a : b; }

// ---------------------------------------------------------------------------
// complex helpers (float2)
// ---------------------------------------------------------------------------
__device__ __forceinline__ float2 cmk(float x, float y) { return make_float2(x, y); }
__device__ __forceinline__ float2 cadd(float2 a, float2 b) { return cmk(a.x + b.x, a.y + b.y); }
__device__ __forceinline__ float2 csub(float2 a, float2 b) { return cmk(a.x - b.x, a.y - b.y); }
__device__ __forceinline__ float2 cmul(float2 a, float2 b) { return cmk(a.x*b.x - a.y*b.y, a.x*b.y + a.y*b.x); }
__device__ __forceinline__ float2 cscale(float c, float2 a) { return cmk(c*a.x, c*a.y); }
__device__ __forceinline__ float2 conjc(float2 a) { return cmk(a.x, -a.y); }
__device__ __forceinline__ float  cabs2(float2 a) { return a.x*a.x + a.y*a.y; }
__device__ __forceinline__ float2 cdiv(float2 a, float2 b) {
    float d = cabs2(b) + 1e-30f;
    return cmk((a.x*b.x + a.y*b.y)/d, (a.y*b.x - a.x*b.y)/d);
}
__device__ __forceinline__ float2 csqrtc(float2 z) {
    float m  = sqrtf(sqrtf(cabs2(z)));
    float th = 0.5f * atan2f(z.y, z.x);
    return cmk(m * cosf(th), m * sinf(th));
}

// ---------------------------------------------------------------------------
// Fused GEMM + bias + optional ReLU using V_WMMA_F32_16X16X4_F32.
// C[M,N] = act(A[M,K] @ W[K,N] + bias[N]).
// Compile-time K/N/NB/RELU: fully unrolled K-loop, branch-free fragment loads
// (clamped address + v_cndmask select for ragged edges), EXEC all-1s at every
// WMMA.  Each wave computes a 16 x (NB*16) strip: one A fragment feeds NB
// B fragments -> NB WMMAs per A load (register blocking for A reuse).
// ---------------------------------------------------------------------------
template <int K, int N, int NB, int RELU>
__global__ __launch_bounds__(256) void hdmd_gemm_wmma(
        const float* __restrict__ A, const float* __restrict__ W,
        const float* __restrict__ bias, float* __restrict__ C, int M)
{
    constexpr int nTilesN = (N + 15) >> 4;
    constexpr int nGroups = (nTilesN + NB - 1) / NB;

    const int wave = blockIdx.x * (blockDim.x >> 5) + (threadIdx.x >> 5);
    const int lane = threadIdx.x & 31;
    const int nTilesM = M >> 4;                  // M is always a multiple of 16 here
    if (wave >= nTilesM * nGroups) return;       // whole wave exits together

    const int tm   = (wave / nGroups) << 4;
    const int tn0  = (wave % nGroups) * (NB << 4);
    const int half = lane >> 4;                  // K-msb half (ISA 16x4 f32 A layout)
    const int lm   = lane & 15;
    const int row  = tm + lm;                    // A fragment row

    int col[NB]; bool cok[NB]; int colc[NB];
#pragma unroll
    for (int j = 0; j < NB; ++j) {
        col[j]  = tn0 + (j << 4) + lm;
        cok[j]  = (N % 16 == 0) ? true : (col[j] < N);
        colc[j] = (N % 16 == 0) ? col[j] : imin(col[j], N - 1);
    }

    v8f acc[NB];
#pragma unroll
    for (int j = 0; j < NB; ++j) acc[j] = v8f{};

    const float* Arow = A + (size_t)row * K;

#pragma unroll
    for (int k0 = 0; k0 < K; k0 += 4) {
        const int kb = k0 + half * 2;            // VGPR0->K=kb, VGPR1->K=kb+1
        v2f a;
        if constexpr (K % 4 == 0) {
            const float2 av = *(const float2*)(Arow + kb);   // global_load_b64
            a.x = av.x; a.y = av.y;
        } else {
            const float ax = Arow[imin(kb, K - 1)];          // clamped, in-bounds
            const float ay = Arow[imin(kb + 1, K - 1)];
            a.x = (kb     < K) ? ax : 0.f;                   // v_cndmask, no branch
            a.y = (kb + 1 < K) ? ay : 0.f;
        }
#pragma unroll
        for (int j = 0; j < NB; ++j) {
            v2f b;
            if constexpr (K % 4 == 0) {
                const float bx = W[(size_t)(kb    ) * N + colc[j]];
                const float by = W[(size_t)(kb + 1) * N + colc[j]];
                b.x = cok[j] ? bx : 0.f;
                b.y = cok[j] ? by : 0.f;
            } else {
                const float bx = W[(size_t)imin(kb,     K - 1) * N + colc[j]];
                const float by = W[(size_t)imin(kb + 1, K - 1) * N + colc[j]];
                b.x = (cok[j] && kb     < K) ? bx : 0.f;
                b.y = (cok[j] && kb + 1 < K) ? by : 0.f;
            }
            acc[j] = __builtin_amdgcn_wmma_f32_16x16x4_f32(
                        /*neg_a=*/false, a, /*neg_b=*/false, b,
                        /*c_mod=*/(short)0, acc[j],
                        /*reuse_a=*/false, /*reuse_b=*/false);
        }
    }

#pragma unroll
    for (int j = 0; j < NB; ++j) {
        if (cok[j]) {
            const float bn = bias[col[j]];
#pragma unroll
            for (int i = 0; i < 8; ++i) {        // VGPR i: M = i (lanes<16) / i+8 (>=16)
                const int m = tm + i + half * 8;
                float v = acc[j][i] + bn;
                if (RELU) v = fmaxf(v, 0.f);
                C[(size_t)m * N + col[j]] = v;
            }
        }
    }
}

// ---------------------------------------------------------------------------
// Hankel-DMD: one wave32 per batch element.
// Rank-31 projected DMD (equivalent nonzero spectrum of the 264x264 operator):
//   G = Ym^T Ym  -> Jacobi eig -> V, sigma
//   Atilde = S^-1 V^T (Ym^T Yp) V S^-1
//   complex Schur of Atilde (Givens Hessenberg + Wilkinson-shift QR)
//   eigenvectors by back-substitution (one per lane), b = w^-1 (S^-1 V^T g0)
//   recon[p,l] = Re( Phi_s[l,:] . (lambda^p * b) )
// ---------------------------------------------------------------------------
__global__ __launch_bounds__(32) void hdmd_dmd_wave(
        const float* __restrict__ ysrc, float* __restrict__ yadv)
{
    const int b    = blockIdx.x;
    const int lane = threadIdx.x;

    __shared__ float  yl[LAT][TT];
    __shared__ float  G[32][32];
    __shared__ float  K2[32][32];
    __shared__ float  V[32][32];
    __shared__ float  At[32][32];
    __shared__ float  Tq[32][32];
    __shared__ float  g0[32], isig[32], y0r[32];
    __shared__ float  M1[LAT][32];
    __shared__ float2 Hc[32][32];
    __shared__ float2 Qc[32][32];
    __shared__ float2 Xc[32][32];
    __shared__ float2 Wv[32][32];
    __shared__ float2 lamv[32], bv[32], zz[32], fcolc[32];
    __shared__ float  cs_r[32];
    __shared__ float2 cs_s[32];
    __shared__ float2 phis[LAT][32];

    // ---- 1. load y for this batch: (T,L) row-major -> yl[l][t]
    for (int idx = lane; idx < LAT * TT; idx += 32) {
        const int t = idx >> 3, l = idx & 7;
        yl[l][t] = ysrc[(size_t)b * (TT * LAT) + t * LAT + l];
    }
    for (int idx = lane; idx < 32 * 32; idx += 32)
        V[idx >> 5][idx & 31] = ((idx >> 5) == (idx & 31)) ? 1.f : 0.f;
    __syncthreads();

    // ---- 2. Gram matrices: G = Ym^T Ym, K2 = Ym^T Yp (Ym(l*33+r, c) = yl[l][r+c])
    for (int idx = lane; idx < NSV * NSV; idx += 32) {
        const int i = idx / NSV, j = idx % NSV;
        float g = 0.f, k2 = 0.f;
        for (int l = 0; l < LAT; ++l)
            for (int r = 0; r < RR; ++r) {
                const float a = yl[l][r + i];
                g  += a * yl[l][r + j];
                k2 += a * yl[l][r + j + 1];
            }
        G[i][j] = g; K2[i][j] = k2;
    }
    __syncthreads();
    if (lane < NSV) g0[lane] = G[lane][0];      // save before Jacobi destroys G
    __syncthreads();

    // ---- 3. cyclic Jacobi on G, accumulate V
    for (int sweep = 0; sweep < 12; ++sweep)
        for (int p = 0; p < NSV - 1; ++p)
            for (int q = p + 1; q < NSV; ++q) {
                const float gpq = G[p][q];
                float c = 1.f, s = 0.f;
                if (fabsf(gpq) > 1e-12f) {
                    const float tau = (G[q][q] - G[p][p]) / (2.f * gpq);
                    const float t   = (tau >= 0.f ? 1.f : -1.f) /
                                      (fabsf(tau) + sqrtf(1.f + tau * tau));
                    c = 1.f / sqrtf(1.f + t * t);
                    s = t * c;
                }
                __syncthreads();
                if (lane < NSV) {
                    const float gp = G[p][lane], gq = G[q][lane];
                    G[p][lane] = c * gp - s * gq;
                    G[q][lane] = s * gp + c * gq;
                }
                __syncthreads();
                if (lane < NSV) {
                    float gp = G[lane][p], gq = G[lane][q];
                    G[lane][p] = c * gp - s * gq;
                    G[lane][q] = s * gp + c * gq;
                    const float vp = V[lane][p], vq = V[lane][q];
                    V[lane][p] = c * vp - s * vq;
                    V[lane][q] = s * vp + c * vq;
                }
                __syncthreads();
            }

    // ---- 4. singular values, reduced initial condition, M1 rows
    if (lane < NSV) isig[lane] = 1.f / sqrtf(fmaxf(G[lane][lane], 1e-20f));
    __syncthreads();
    if (lane < NSV) {
        float sm = 0.f;
        for (int k = 0; k < NSV; ++k) sm += V[k][lane] * g0[k];
        y0r[lane] = sm * isig[lane];
    }
    for (int idx = lane; idx < NSV * NSV; idx += 32) {      // Tq = K2 @ V
        const int i = idx / NSV, j = idx % NSV;
        float sm = 0.f;
        for (int k = 0; k < NSV; ++k) sm += K2[i][k] * V[k][j];
        Tq[i][j] = sm;
    }
    __syncthreads();
    for (int idx = lane; idx < NSV * NSV; idx += 32) {      // At = S^-1 V^T Tq S^-1
        const int i = idx / NSV, j = idx % NSV;
        float sm = 0.f;
        for (int k = 0; k < NSV; ++k) sm += V[k][i] * Tq[k][j];
        At[i][j] = sm * isig[i] * isig[j];
    }
    for (int idx = lane; idx < LAT * NSV; idx += 32) {      // M1[l] = yl[l][0:31] V S^-1
        const int l = idx / NSV, i = idx % NSV;
        float sm = 0.f;
        for (int c = 0; c < NSV; ++c) sm += yl[l][c] * V[c][i];
        M1[l][i] = sm * isig[i];
    }
    __syncthreads();

    // ---- 5. Givens Hessenberg reduction of At, accumulate Q in Tq
    for (int idx = lane; idx < 32 * 32; idx += 32)
        Tq[idx >> 5][idx & 31] = ((idx >> 5) == (idx & 31)) ? 1.f : 0.f;
    __syncthreads();
    for (int j = 0; j < NSV - 2; ++j)
        for (int i = j + 2; i < NSV; ++i) {
            const float a = At[j + 1][j], bq = At[i][j];
            const float r = sqrtf(a * a + bq * bq);
            float c = 1.f, s = 0.f;
            if (r > 1e-20f) { c = a / r; s = bq / r; }
            __syncthreads();
            if (lane < NSV) {
                const float r1 = At[j + 1][lane], r2 = At[i][lane];
                At[j + 1][lane] =  c * r1 + s * r2;
                At[i][lane]     = -s * r1 + c * r2;
            }
            __syncthreads();
            if (lane < NSV) {
                const float c1 = At[lane][j + 1], c2 = At[lane][i];
                At[lane][j + 1] =  c * c1 + s * c2;
                At[lane][i]     = -s * c1 + c * c2;
                const float q1 = Tq[lane][j + 1], q2 = Tq[lane][i];
                Tq[lane][j + 1] =  c * q1 + s * q2;
                Tq[lane][i]     = -s * q1 + c * q2;
            }
            __syncthreads();
        }
    for (int idx = lane; idx < NSV * NSV; idx += 32) {
        const int i = idx / NSV, j = idx % NSV;
        Hc[i][j] = cmk(At[i][j], 0.f);
        Qc[i][j] = cmk(Tq[i][j], 0.f);
    }
    __syncthreads();

    // ---- 6. complex Wilkinson-shift QR with bottom deflation
    int m = NSV - 1;
    for (int iter = 0; iter < 1500 && m > 0; ++iter) {
        const float sub = sqrtf(cabs2(Hc[m][m - 1]));
        const float dia = sqrtf(cabs2(Hc[m - 1][m - 1])) + sqrtf(cabs2(Hc[m][m]));
        if (sub <= 1e-7f * (dia + 1e-20f)) {
            if (lane == 0) Hc[m][m - 1] = cmk(0.f, 0.f);
            __syncthreads();
            --m; continue;
        }
        const float2 aa = Hc[m - 1][m - 1], bq = Hc[m - 1][m];
        const float2 cq = Hc[m][m - 1],     dd = Hc[m][m];
        const float2 delta = cmk(0.5f * (aa.x - dd.x), 0.5f * (aa.y - dd.y));
        const float2 bc    = cmul(bq, cq);
        const float2 disc  = csqrtc(cadd(cmul(delta, delta), bc));
        const float2 dp = cadd(delta, disc), dm = csub(delta, disc);
        const float2 den = (cabs2(dp) > cabs2(dm)) ? dp : dm;
        float2 mu = dd;
        if (cabs2(den) > 1e-30f) mu = csub(dd, cdiv(bc, den));
        __syncthreads();
        if (lane <= m) Hc[lane][lane] = csub(Hc[lane][lane], mu);
        __syncthreads();
        for (int k = 0; k < m; ++k) {                       // QR sweep (rows)
            const float2 ha = Hc[k][k], hb = Hc[k + 1][k];
            const float na = sqrtf(cabs2(ha));
            const float tt = sqrtf(cabs2(ha) + cabs2(hb));
            float cc; float2 ss;
            if (tt < 1e-20f)      { cc = 1.f; ss = cmk(0.f, 0.f); }
            else if (na < 1e-20f) { cc = 0.f; ss = cmk(hb.x / tt, -hb.y / tt); }
            else {
                cc = na / tt;
                const float2 an = cmk(ha.x / na, ha.y / na);
                ss = cscale(1.f / tt, cmul(an, conjc(hb)));
            }
            if (lane == 0) { cs_r[k] = cc; cs_s[k] = ss; }
            __syncthreads();
            if (lane < NSV && lane >= k) {
                const float2 x1 = Hc[k][lane], x2 = Hc[k + 1][lane];
                Hc[k][lane]     = cadd(cscale(cc, x1), cmul(ss, x2));
                Hc[k + 1][lane] = csub(cscale(cc, x2), cmul(conjc(ss), x1));
            }
            __syncthreads();
        }
        for (int k = 0; k < m; ++k) {                       // RQ (columns) + accumulate Q
            const float cc = cs_r[k]; const float2 ss = cs_s[k];
            if (lane < NSV) {
                const float2 x1 = Hc[lane][k], x2 = Hc[lane][k + 1];
                Hc[lane][k]     = cadd(cscale(cc, x1), cmul(conjc(ss), x2));
                Hc[lane][k + 1] = csub(cscale(cc, x2), cmul(ss, x1));
                const float2 q1 = Qc[lane][k], q2 = Qc[lane][k + 1];
                Qc[lane][k]     = cadd(cscale(cc, q1), cmul(conjc(ss), q2));
                Qc[lane][k + 1] = csub(cscale(cc, q2), cmul(ss, q1));
            }
            __syncthreads();
        }
        if (lane <= m) Hc[lane][lane] = cadd(Hc[lane][lane], mu);
        __syncthreads();
    }
    if (lane < NSV) lamv[lane] = Hc[lane][lane];
    __syncthreads();

    // ---- 7. eigenvectors of triangular T: one eigenvector per lane
    if (lane < NSV) {
        const int k = lane;
        for (int i = 0; i < NSV; ++i) Xc[i][k] = cmk(0.f, 0.f);
        Xc[k][k] = cmk(1.f, 0.f);
        const float2 lam = lamv[k];
        for (int i = k - 1; i >= 0; --i) {
            float2 sm = cmk(0.f, 0.f);
            for (int j = i + 1; j <= k; ++j) sm = cadd(sm, cmul(Hc[i][j], Xc[j][k]));
            float2 dn = csub(lam, Hc[i][i]);
            if (cabs2(dn) < 1e-16f) dn = cmk(1e-8f, 0.f);
            Xc[i][k] = cdiv(sm, dn);
        }
    }
    __syncthreads();
    if (lane < NSV) {                                       // Wv = Q X, normalized
        const int k = lane;
        float nrm = 0.f;
        for (int i = 0; i < NSV; ++i) {
            float2 sm = cmk(0.f, 0.f);
            for (int j = 0; j <= k; ++j) sm = cadd(sm, cmul(Qc[i][j], Xc[j][k]));
            Wv[i][k] = sm; nrm += cabs2(sm);
        }
        const float inv = 1.f / sqrtf(nrm + 1e-30f);
        for (int i = 0; i < NSV; ++i) Wv[i][k] = cscale(inv, Wv[i][k]);
    }
    __syncthreads();

    // ---- 8. amplitudes: solve Wv b = y0r (complex LU w/ partial pivoting in Qc)
    for (int idx = lane; idx < NSV * NSV; idx += 32)
        Qc[idx / NSV][idx % NSV] = Wv[idx / NSV][idx % NSV];
    if (lane < NSV) bv[lane] = cmk(y0r[lane], 0.f);
    __syncthreads();
    for (int k = 0; k < NSV; ++k) {
        int piv = k; float best = cabs2(Qc[k][k]);
        for (int i = k + 1; i < NSV; ++i) {
            const float v = cabs2(Qc[i][k]);
            if (v > best) { best = v; piv = i; }
        }
        if (piv != k) {
            if (lane < NSV) { const float2 t = Qc[k][lane]; Qc[k][lane] = Qc[piv][lane]; Qc[piv][lane] = t; }
            if (lane == 0)  { const float2 t = bv[k]; bv[k] = bv[piv]; bv[piv] = t; }
        }
        __syncthreads();
        float2 pv = Qc[k][k];
        if (cabs2(pv) < 1e-30f) pv = cmk(1e-15f, 0.f);
        if (lane < NSV && lane > k) fcolc[lane] = cdiv(Qc[lane][k], pv);
        __syncthreads();
        if (lane < NSV && lane >= k)
            for (int i = k + 1; i < NSV; ++i)
                Qc[i][lane] = csub(Qc[i][lane], cmul(fcolc[i], Qc[k][lane]));
        if (lane == 0)
            for (int i = k + 1; i < NSV; ++i)
                bv[i] = csub(bv[i], cmul(fcolc[i], bv[k]));
        __syncthreads();
    }
    if (lane == 0) {
        for (int i = NSV - 1; i >= 0; --i) {
            float2 sm = bv[i];
            for (int j = i + 1; j < NSV; ++j) sm = csub(sm, cmul(Qc[i][j], bv[j]));
            float2 dn = Qc[i][i];
            if (cabs2(dn) < 1e-30f) dn = cmk(1e-15f, 0.f);
            bv[i] = cdiv(sm, dn);
        }
    }
    __syncthreads();

    // ---- 9. sampled modes Phi_s[l] = M1[l] . Wv, then rollout lambda^p b
    for (int idx = lane; idx < LAT * NSV; idx += 32) {
        const int l = idx / NSV, mm = idx % NSV;
        float2 sm = cmk(0.f, 0.f);
        for (int k = 0; k < NSV; ++k) {
            const float2 w = Wv[k][mm];
            sm.x += M1[l][k] * w.x; sm.y += M1[l][k] * w.y;
        }
        phis[l][mm] = sm;
    }
    if (lane < NSV) zz[lane] = bv[lane];
    __syncthreads();

    float* out = yadv + (size_t)b * (PRED * LAT);
    for (int p = 0; p < PRED; ++p) {
        if (lane < LAT) {
            float acc = 0.f;
            for (int mm = 0; mm < NSV; ++mm)
                acc += phis[lane][mm].x * zz[mm].x - phis[lane][mm].y * zz[mm].y;
            out[p * LAT + lane] = acc;
        }
        __syncthreads();
        if (lane < NSV) zz[lane] = cmul(zz[lane], lamv[lane]);
        __syncthreads();
    }
}

// ---------------------------------------------------------------------------
// Host-side orchestration
// ---------------------------------------------------------------------------
static inline unsigned hdmd_nblocks(int M, int N, int NB) {
    const int nTilesN = (N + 15) >> 4;
    const int nGroups = (nTilesN + NB - 1) / NB;
    const int waves   = (M >> 4) * nGroups;
    return (unsigned)((waves + 7) / 8);              // 8 waves / 256-thread block
}

extern "C" void kernel_launch(void* const* d_in, const int* in_sizes, int n_in,
                              void* d_out, int out_size, void* d_ws, size_t ws_size,
                              hipStream_t stream)
{
    (void)in_sizes; (void)n_in; (void)out_size; (void)ws_size;

    const float* x         = (const float*)d_in[0];
    const float* enc_w_in  = (const float*)d_in[1];
    const float* enc_b_in  = (const float*)d_in[2];
    const float* enc_w_h   = (const float*)d_in[3];
    const float* enc_b_h   = (const float*)d_in[4];
    const float* enc_w_out = (const float*)d_in[5];
    const float* enc_b_out = (const float*)d_in[6];
    const float* dec_w_in  = (const float*)d_in[7];
    const float* dec_b_in  = (const float*)d_in[8];
    const float* dec_w_h   = (const float*)d_in[9];
    const float* dec_b_h   = (const float*)d_in[10];
    const float* dec_w_out = (const float*)d_in[11];
    const float* dec_b_out = (const float*)d_in[12];

    float* out   = (float*)d_out;
    float* y     = out;                                   // (B,T,LAT)    131072
    float* x_ae  = out + MROWS * LAT;                     // (B,T,PHYS)    49152
    float* x_adv = x_ae + MROWS * PHYS;                   // (B,T,PHYS)    49152
    float* y_adv = x_adv + MROWS * PHYS;                  // (B,PRED,LAT) 131072

    float* ws0 = (float*)d_ws;
    float* ws1 = ws0 + (size_t)MROWS * NN;

    const dim3 blk(256);
    const unsigned gb_wide = hdmd_nblocks(MROWS, NN, 4);   // N=128, NB=4
    const unsigned gb_thin = hdmd_nblocks(MROWS, 16, 1);   // N<=16, NB=1

    // encoder: x -> ws0 -> ws1 -> ws0 -> ws1 -> y
    hdmd_gemm_wmma<PHYS, NN, 4, 1><<<gb_wide, blk, 0, stream>>>(x, enc_w_in, enc_b_in, ws0, MROWS);
    hdmd_gemm_wmma<NN, NN, 4, 1><<<gb_wide, blk, 0, stream>>>(ws0, enc_w_h + 0 * NN * NN, enc_b_h + 0 * NN, ws1, MROWS);
    hdmd_gemm_wmma<NN, NN, 4, 1><<<gb_wide, blk, 0, stream>>>(ws1, enc_w_h + 1 * NN * NN, enc_b_h + 1 * NN, ws0, MROWS);
    hdmd_gemm_wmma<NN, NN, 4, 1><<<gb_wide, blk, 0, stream>>>(ws0, enc_w_h + 2 * NN * NN, enc_b_h + 2 * NN, ws1, MROWS);
    hdmd_gemm_wmma<NN, LAT, 1, 0><<<gb_thin, blk, 0, stream>>>(ws1, enc_w_out, enc_b_out, y, MROWS);

    // decoder on y -> x_ae
    hdmd_gemm_wmma<LAT, NN, 4, 1><<<gb_wide, blk, 0, stream>>>(y, dec_w_in, dec_b_in, ws0, MROWS);
    hdmd_gemm_wmma<NN, NN, 4, 1><<<gb_wide, blk, 0, stream>>>(ws0, dec_w_h + 0 * NN * NN, dec_b_h + 0 * NN, ws1, MROWS);
    hdmd_gemm_wmma<NN, NN, 4, 1><<<gb_wide, blk, 0, stream>>>(ws1, dec_w_h + 1 * NN * NN, dec_b_h + 1 * NN, ws0, MROWS);
    hdmd_gemm_wmma<NN, NN, 4, 1><<<gb_wide, blk, 0, stream>>>(ws0, dec_w_h + 2 * NN * NN, dec_b_h + 2 * NN, ws1, MROWS);
    hdmd_gemm_wmma<NN, PHYS, 1, 0><<<gb_thin, blk, 0, stream>>>(ws1, dec_w_out, dec_b_out, x_ae, MROWS);

    // Hankel-DMD: one wave per batch element
    hdmd_dmd_wave<<<BB, 32, 0, stream>>>(y, y_adv);

    // decoder on y_adv -> x_adv
    hdmd_gemm_wmma<LAT, NN, 4, 1><<<gb_wide, blk, 0, stream>>>(y_adv, dec_w_in, dec_b_in, ws0, MROWS);
    hdmd_gemm_wmma<NN, NN, 4, 1><<<gb_wide, blk, 0, stream>>>(ws0, dec_w_h + 0 * NN * NN, dec_b_h + 0 * NN, ws1, MROWS);
    hdmd_gemm_wmma<NN, NN, 4, 1><<<gb_wide, blk, 0, stream>>>(ws1, dec_w_h + 1 * NN * NN, dec_b_h + 1 * NN, ws0, MROWS);
    hdmd_gemm_wmma<NN, NN, 4, 1><<<gb_wide, blk, 0, stream>>>(ws0, dec_w_h + 2 * NN * NN, dec_b_h + 2 * NN, ws1, MROWS);
    hdmd_gemm_wmma<NN, PHYS, 1, 0><<<gb_thin, blk, 0, stream>>>(ws1, dec_w_out, dec_b_out, x_adv, MROWS);
}